// LSSL_9680856285946
// MI455X (gfx1250) — compile-verified
//
#include <hip/hip_runtime.h>
#include <stdint.h>

typedef __attribute__((ext_vector_type(2))) float v2f;
typedef __attribute__((ext_vector_type(8))) float v8f;

#define BSZ     8
#define SEQ     4096
#define NCHN    1024
#define BLOCK_N 128          // channels per block
#define ROWS    64           // sequence rows per staged chunk
#define NCHUNK  8            // chunks per superchunk
#define SUPER   (ROWS * NCHUNK)  // 512 sequence steps per block
#define THREADS 256

__global__ __launch_bounds__(THREADS) void lssl_zero(float* out) {
    int i = blockIdx.x * blockDim.x + threadIdx.x;
    if (i < BSZ * NCHN) out[i] = 0.0f;
}

__global__ __launch_bounds__(THREADS) void lssl_main(
    const float* __restrict__ x, const float* __restrict__ Amat,
    const float* __restrict__ Bvec, const float* __restrict__ Cvec,
    float* __restrict__ out)
{
    // double-buffered staging tile: 2 x (64 rows x 128 channels) f32 = 64 KB
    __shared__ float smem[2 * ROWS * BLOCK_N];

    // *** Escape hatch: capture the LDS allocation so the compiler must assume
    // the "memory"-clobbering async-copy asm below writes it. Without this the
    // LDS array (written only by tensor/async hardware, not visible stores)
    // gets folded to undef and deleted. Emits zero instructions.
    {
        float* lds_escape = smem;
        asm volatile("" :: "v"(lds_escape) : "memory");
    }

    const int tid  = threadIdx.x;
    const int lane = tid & 31;
    const int wave = tid >> 5;

    const int bi   = blockIdx.x;
    const int b    = bi >> 6;          // 0..7
    const int tile = (bi >> 3) & 7;    // 0..7  (128-channel tile)
    const int sc   = bi & 7;           // 0..7  (512-step superchunk)
    const int nb0  = tile * BLOCK_N;
    const int l0   = sc * SUPER;

    // saddr carries batch + channel-tile base; voffset carries (row,col)
    const float* xbase = x + (size_t)b * SEQ * NCHN + nb0;

    // ---- per-lane channel & WMMA row mapping -------------------------------
    // lanes 0..15  -> channel m = lane,    supply K = {0,1} (rows 4t, 4t+1)
    // lanes 16..31 -> channel m = lane-16, supply K = {2,3} (rows 4t+2, 4t+3)
    const int m    = lane & 15;
    const int kodd = lane >> 4;
    const int c    = nb0 + wave * 16 + m;

    const float dt   = 1e-3f;
    const float a    = Amat[(size_t)c * NCHN + c];      // diag(A)
    const float bv   = Bvec[c];
    const float bd   = (1.0f + dt * bv) / (1.0f - dt * bv);
    const float base = Cvec[c] * bd;

    const float a2 = a * a,  a4 = a2 * a2;
    const float a8 = a4 * a4, a16 = a8 * a8, a32 = a16 * a16, a64 = a32 * a32;

    // starting weight for this lane's first tap: base * a^(l0 + 2*kodd)
    const float e0 = (float)(l0 + 2 * kodd);
    float wchunk = base * __powf(fmaxf(a, 1e-38f), e0);  // branchless; e0==0 -> 1

    // ---- async-copy geometry ----------------------------------------------
    // one chunk = 64 rows x 512 B; 256 threads x 16 B -> 8 sweeps of 8 rows
    const int crow = tid >> 5;            // row within a sweep (0..7)
    const int ccol = (tid & 31) * 4;      // float column (0..124)

    auto issue_chunk = [&](int ch, int buf) {
        const unsigned lbase = (unsigned)buf * (ROWS * BLOCK_N * 4u);
        const unsigned grow0 = (unsigned)(l0 + ch * ROWS);
        #pragma unroll
        for (int s = 0; s < 8; ++s) {
            const int r = s * 8 + crow;
            const unsigned voff = ((grow0 + (unsigned)r) * NCHN + (unsigned)ccol) * 4u;
            const unsigned loff = lbase + (unsigned)(r * BLOCK_N + ccol) * 4u;
            asm volatile("global_load_async_to_lds_b128 %0, %1, %2"
                         :: "v"(loff), "v"(voff), "s"(xbase)
                         : "memory");
        }
    };

    v8f acc = {0.f, 0.f, 0.f, 0.f, 0.f, 0.f, 0.f, 0.f};
    v2f ones2; ones2.x = 1.0f; ones2.y = 1.0f;   // B = all-ones -> WMMA does row-sum

    issue_chunk(0, 0);

    const int cl = wave * 16 + m;                 // channel column inside LDS tile
    for (int ch = 0; ch < NCHUNK; ++ch) {
        const int buf = ch & 1;
        if (ch + 1 < NCHUNK) {
            issue_chunk(ch + 1, (ch + 1) & 1);
            asm volatile("s_wait_asynccnt 0x8" ::: "memory");  // chunk ch complete
        } else {
            asm volatile("s_wait_asynccnt 0x0" ::: "memory");
        }
        __syncthreads();

        const float* sb = smem + buf * (ROWS * BLOCK_N);
        float wa = wchunk;
        #pragma unroll 4
        for (int t = 0; t < ROWS / 4; ++t) {
            const int r0 = 4 * t + 2 * kodd;
            const float x0 = sb[r0 * BLOCK_N + cl];
            const float x1 = sb[(r0 + 1) * BLOCK_N + cl];
            v2f af;
            af.x = x0 * wa;
            af.y = x1 * (wa * a);
            // D[m,n] += sum_k A[m,k] * 1  -> per-channel partial sums in every column
            acc = __builtin_amdgcn_wmma_f32_16x16x4_f32(
                      false, af, false, ones2, (short)0, acc, false, false);
            wa *= a4;
        }
        wchunk *= a64;
        __syncthreads();
    }

    // D layout (f32 16x16): VGPR i, lanes 0-15 hold D[i, lane]; lanes 16-31 hold D[8+i, lane-16].
    // Columns are identical, so lane 0 carries channels m=0..7, lane 16 carries m=8..15.
    if ((lane & 15) == 0) {
        float* ob = out + b * NCHN + (nb0 + wave * 16 + kodd * 8);
        atomicAdd(ob + 0, acc[0]);
        atomicAdd(ob + 1, acc[1]);
        atomicAdd(ob + 2, acc[2]);
        atomicAdd(ob + 3, acc[3]);
        atomicAdd(ob + 4, acc[4]);
        atomicAdd(ob + 5, acc[5]);
        atomicAdd(ob + 6, acc[6]);
        atomicAdd(ob + 7, acc[7]);
    }
}

extern "C" void kernel_launch(void* const* d_in, const int* in_sizes, int n_in,
                              void* d_out, int out_size, void* d_ws, size_t ws_size,
                              hipStream_t stream) {
    const float* x = (const float*)d_in[0];
    const float* A = (const float*)d_in[1];
    const float* B = (const float*)d_in[2];
    const float* C = (const float*)d_in[3];
    float* out = (float*)d_out;

    lssl_zero<<<(BSZ * NCHN + THREADS - 1) / THREADS, THREADS, 0, stream>>>(out);

    const int nblocks = BSZ * (NCHN / BLOCK_N) * (SEQ / SUPER);  // 8*8*8 = 512
    lssl_main<<<nblocks, THREADS, 0, stream>>>(x, A, B, C, out);
}